// DotLoss_50714973831542
// MI455X (gfx1250) — compile-verified
//
#include <hip/hip_runtime.h>

// DotLoss on MI455X (gfx1250): memory-bound (~256 MB @ 23.3 TB/s ≈ 11 us floor).
// Row-wise dot products computed as diagonals of V_WMMA_F32_16X16X4_F32 tiles.
//
// v4: the compiler schedules loads with a full s_wait_loadcnt 0 drain (1-2
// loads in flight per wave), so latency hiding must come from wave count.
// Split K 4-ways: 4 waves cooperate on each 16-row tile (128 columns each),
// combine partial diagonals through LDS, one wave applies the hinge. 8192
// waves total -> ~4x the aggregate outstanding-load bytes, zero extra HBM
// traffic, same WMMA diagonal trick.

typedef __attribute__((ext_vector_type(2))) float v2f;
typedef __attribute__((ext_vector_type(4))) float v4f;
typedef __attribute__((ext_vector_type(8))) float v8f;

#define N_ROWS 32768
#define DIM    512
#define KSPLIT 4                    // waves per tile (K quarters)
#define TILES_PER_BLOCK 2
#define WAVES_PER_BLOCK (KSPLIT * TILES_PER_BLOCK)   // 8 waves = 256 threads
#define KBATCH 4                    // column-groups of 8 batched per load group

__global__ void dotloss_zero_kernel(float* out) {
    if (threadIdx.x == 0) out[0] = 0.0f;
}

__device__ __forceinline__ v2f vlo(v4f x) { return __builtin_shufflevector(x, x, 0, 1); }
__device__ __forceinline__ v2f vhi(v4f x) { return __builtin_shufflevector(x, x, 2, 3); }

__global__ __launch_bounds__(32 * WAVES_PER_BLOCK)
void dotloss_wmma_kernel(const float* __restrict__ img,
                         const float* __restrict__ aud,
                         const int*   __restrict__ Iidx,
                         const int*   __restrict__ Aidx,
                         float*       __restrict__ out) {
    const int lane  = threadIdx.x & 31;
    const int wave  = threadIdx.x >> 5;
    const int tib   = wave >> 2;          // tile within block (0..1)
    const int q     = wave & (KSPLIT - 1);// K-quarter handled by this wave
    const int tile  = blockIdx.x * TILES_PER_BLOCK + tib;
    const int m     = lane & 15;          // row within tile
    const int chalf = (lane >> 4) << 2;   // 0 or 4 (column sub-group)
    const int row   = tile * 16 + m;

    // Identical per-lane addressing serves both the A (16x4, M-major) and
    // B (4x16, N-major == transposed audio) fragment layouts.
    const float* pImg  = img + (size_t)row * DIM + chalf;
    const float* pAud  = aud + (size_t)row * DIM + chalf;
    const float* pImgI = img + (size_t)Iidx[row] * DIM + chalf;
    const float* pAudA = aud + (size_t)Aidx[row] * DIM + chalf;

    v8f c_an = {};  // anchor:  image[i]        . audio[i]
    v8f c_ii = {};  // Iimpsim: image[I_imp[i]] . audio[i]
    v8f c_ai = {};  // Aimpsim: image[i]        . audio[A_imp[i]]

    const int kbase = q * (DIM / KSPLIT);          // 128-column quarter
    const int kend  = kbase + (DIM / KSPLIT);

#pragma unroll
    for (int kk = kbase; kk < kend; kk += 8 * KBATCH) {
        v4f a[KBATCH], b[KBATCH], ag[KBATCH], bg[KBATCH];
        // Issue all b128 loads first -> deep outstanding-load window.
#pragma unroll
        for (int j = 0; j < KBATCH; ++j) {
            const int o = kk + 8 * j;
            a[j]  = *(const v4f*)(pImg  + o);
            b[j]  = *(const v4f*)(pAud  + o);
            ag[j] = *(const v4f*)(pImgI + o);
            bg[j] = *(const v4f*)(pAudA + o);
        }
        // Two WMMA k-steps per b128: pairs {0,1} then {2,3} of each load.
        // (Diagonal is invariant to k-slot permutation as long as A and B
        // use the same permutation.)
#pragma unroll
        for (int j = 0; j < KBATCH; ++j) {
            c_an = __builtin_amdgcn_wmma_f32_16x16x4_f32(false, vlo(a[j]),  false, vlo(b[j]),
                                                         (short)0, c_an, false, false);
            c_ii = __builtin_amdgcn_wmma_f32_16x16x4_f32(false, vlo(ag[j]), false, vlo(b[j]),
                                                         (short)0, c_ii, false, false);
            c_ai = __builtin_amdgcn_wmma_f32_16x16x4_f32(false, vlo(a[j]),  false, vlo(bg[j]),
                                                         (short)0, c_ai, false, false);
            c_an = __builtin_amdgcn_wmma_f32_16x16x4_f32(false, vhi(a[j]),  false, vhi(b[j]),
                                                         (short)0, c_an, false, false);
            c_ii = __builtin_amdgcn_wmma_f32_16x16x4_f32(false, vhi(ag[j]), false, vhi(b[j]),
                                                         (short)0, c_ii, false, false);
            c_ai = __builtin_amdgcn_wmma_f32_16x16x4_f32(false, vhi(a[j]),  false, vhi(bg[j]),
                                                         (short)0, c_ai, false, false);
        }
    }

    // Diagonal extraction: element (i,i) sits in VGPR i%8 of lane (i<8 ? i : i+16).
    float an = 0.0f, ii = 0.0f, ai = 0.0f;
#pragma unroll
    for (int r = 0; r < 8; ++r) {
        if (lane == r || lane == r + 24) {
            an = c_an[r]; ii = c_ii[r]; ai = c_ai[r];
        }
    }

    // Cross-wave (K-quarter) combine through LDS.
    __shared__ float red[TILES_PER_BLOCK][KSPLIT][3][16];
    const bool diagLane = (lane < 8) || (lane >= 24);
    const int  i        = (lane < 8) ? lane : (lane - 16);
    if (diagLane) {
        red[tib][q][0][i] = an;
        red[tib][q][1][i] = ii;
        red[tib][q][2][i] = ai;
    }
    __syncthreads();

    if (q == 0) {
        float v = 0.0f;
        if (lane < 16) {
            float An = 0.0f, Ii = 0.0f, Ai = 0.0f;
#pragma unroll
            for (int w = 0; w < KSPLIT; ++w) {
                An += red[tib][w][0][lane];
                Ii += red[tib][w][1][lane];
                Ai += red[tib][w][2][lane];
            }
            float t1 = 1.0f + Ii - An;   // MARGIN = 1.0
            float t2 = 1.0f + Ai - An;
            v = fmaxf(t1, 0.0f) + fmaxf(t2, 0.0f);
        }
        // wave32 tree reduction
#pragma unroll
        for (int off = 16; off > 0; off >>= 1)
            v += __shfl_xor(v, off, 32);
        if (lane == 0)
            atomicAdd(out, v * (1.0f / (float)N_ROWS));
    }
}

extern "C" void kernel_launch(void* const* d_in, const int* in_sizes, int n_in,
                              void* d_out, int out_size, void* d_ws, size_t ws_size,
                              hipStream_t stream) {
    const float* img  = (const float*)d_in[0];
    const float* aud  = (const float*)d_in[1];
    const int*   Iidx = (const int*)d_in[2];
    const int*   Aidx = (const int*)d_in[3];
    float* out = (float*)d_out;

    dotloss_zero_kernel<<<1, 32, 0, stream>>>(out);

    const int tiles  = N_ROWS / 16;                    // 2048 tiles, 8192 waves
    const int blocks = tiles / TILES_PER_BLOCK;        // 1024 blocks of 256 threads
    dotloss_wmma_kernel<<<blocks, 32 * WAVES_PER_BLOCK, 0, stream>>>(
        img, aud, Iidx, Aidx, out);
}